// Wrapper_59150289601023
// MI455X (gfx1250) — compile-verified
//
#include <hip/hip_runtime.h>
#include <stdint.h>

#define C_CLS   80
#define A_NUM   9
#define BATCH   8
#define TOPK    1000
#define MAXOUT  100
#define CAND_CAP 4096
#define NBINS   4096             // 12-bit ordered-key bins
#define PAD     1024
#define NTILE   64               // PAD/16
#define NPAIRTILES 2080          // 64*65/2
#define CONF_THR 0.05f
#define IOU_THR  0.5f
#define CLS_OFFSET 10000.0f
#define TOT4_PER_B 1534500       // 6,138,000 / 4 float4s per batch

typedef float v2f __attribute__((ext_vector_type(2)));
typedef float v8f __attribute__((ext_vector_type(8)));

// ---------- ws layout (bytes) ----------
#define WS_HIST    0            // u32[8*4096]               131,072
#define WS_MASK    131072       // u32[8*1024*32]          1,048,576
#define WS_CCNT    1179648      // u32[8]
#define WS_THR     1179680      // u32[8]
#define WS_CAND    1179712      // u32[8*4096*2]             262,144
#define WS_TSC     1441856      // f32[8*1000]
#define WS_TIDX    1473856      // u32[8*1000]
#define WS_BOX     1505856      // f32[8*1000*4]
#define WS_LAB     1633856      // i32[8*1000]
#define WS_ZERO_WORDS 294928    // hist+mask+ccnt+thr words to clear

__device__ __forceinline__ uint32_t orderKey(float x) {
  uint32_t u = __float_as_uint(x);
  return (u & 0x80000000u) ? ~u : (u | 0x80000000u);
}
__device__ __forceinline__ float keyToFloat(uint32_t k) {
  uint32_t u = (k & 0x80000000u) ? (k & 0x7FFFFFFFu) : ~k;
  return __uint_as_float(u);
}

// map per-batch float4 index -> (level ptr, r4 within batch chunk, spatial, lvl anchor off)
__device__ __forceinline__ void lvlMap4(int v, const float* l0, const float* l1,
                                        const float* l2, const float* l3, const float* l4,
                                        const float*& p, int& r4, int& s, int& lvlOff) {
  if      (v < 1152000) { p = l0; r4 = v;           s = 6400; lvlOff = 0;     }
  else if (v < 1440000) { p = l1; r4 = v - 1152000; s = 1600; lvlOff = 57600; }
  else if (v < 1512000) { p = l2; r4 = v - 1440000; s = 400;  lvlOff = 72000; }
  else if (v < 1530000) { p = l3; r4 = v - 1512000; s = 100;  lvlOff = 75600; }
  else                  { p = l4; r4 = v - 1530000; s = 25;   lvlOff = 76500; }
}

__global__ void zero_kernel(uint32_t* p, int n) {
  int stride = gridDim.x * blockDim.x;
  for (int i = blockIdx.x * blockDim.x + threadIdx.x; i < n; i += stride) p[i] = 0u;
}

// ---- pass 1: LDS-aggregated 12-bit histogram, one batch per blockIdx.y ----
__global__ void __launch_bounds__(256) hist_kernel(const float* l0, const float* l1,
                                                   const float* l2, const float* l3,
                                                   const float* l4, uint32_t* hist) {
  __shared__ uint32_t lh[NBINS];        // 16 KB of the 320 KB/WGP LDS
  for (int i = threadIdx.x; i < NBINS; i += blockDim.x) lh[i] = 0u;
  __syncthreads();
  int b = blockIdx.y;
  int stride = gridDim.x * blockDim.x;
  for (int v = blockIdx.x * blockDim.x + threadIdx.x; v < TOT4_PER_B; v += stride) {
    const float* p; int r4, s, lvlOff;
    lvlMap4(v, l0, l1, l2, l3, l4, p, r4, s, lvlOff);
    const float4* p4 = (const float4*)(p + (size_t)b * (A_NUM * C_CLS) * s);
    __builtin_prefetch(p4 + r4 + stride, 0, 1);      // global_prefetch_b8
    float4 x = p4[r4];
    atomicAdd(&lh[orderKey(x.x) >> 20], 1u);
    atomicAdd(&lh[orderKey(x.y) >> 20], 1u);
    atomicAdd(&lh[orderKey(x.z) >> 20], 1u);
    atomicAdd(&lh[orderKey(x.w) >> 20], 1u);
  }
  __syncthreads();
  for (int i = threadIdx.x; i < NBINS; i += blockDim.x) {
    uint32_t c = lh[i];
    if (c) atomicAdd(&hist[b * NBINS + i], c);
  }
}

// ---- pass 2: find threshold bin for the 1000th largest value ----
__global__ void scan_kernel(const uint32_t* hist, uint32_t* thrKey) {
  int b = threadIdx.x;
  if (b >= BATCH) return;
  const uint32_t* h = hist + b * NBINS;
  uint32_t acc = 0; int thr = 0;
  for (int bin = NBINS - 1; bin >= 0; bin--) {
    acc += h[bin];
    if (acc >= TOPK) { thr = bin; break; }
  }
  thrKey[b] = (uint32_t)thr << 20;
}

// ---- pass 3: collect candidates >= threshold with flat score index ----
__global__ void __launch_bounds__(256) collect_kernel(const float* l0, const float* l1,
                                                      const float* l2, const float* l3,
                                                      const float* l4, const uint32_t* thrKey,
                                                      uint32_t* candCnt, uint32_t* cand) {
  int b = blockIdx.y;
  uint32_t thr = thrKey[b];
  int stride = gridDim.x * blockDim.x;
  for (int v = blockIdx.x * blockDim.x + threadIdx.x; v < TOT4_PER_B; v += stride) {
    const float* p; int r4, s, lvlOff;
    lvlMap4(v, l0, l1, l2, l3, l4, p, r4, s, lvlOff);
    const float4* p4 = (const float4*)(p + (size_t)b * (A_NUM * C_CLS) * s);
    float4 x = p4[r4];
    float e[4] = { x.x, x.y, x.z, x.w };
    #pragma unroll
    for (int q = 0; q < 4; q++) {
      uint32_t key = orderKey(e[q]);
      if (key >= thr) {
        int r  = r4 * 4 + q;          // element index within [720][s] batch chunk
        int ch = r / s;               // a*C + c
        int sp = r % s;
        int a  = ch / C_CLS;
        int c  = ch % C_CLS;
        int anchor = lvlOff + sp * A_NUM + a;
        uint32_t flat = (uint32_t)(anchor * C_CLS + c);
        uint32_t pos = atomicAdd(&candCnt[b], 1u);
        if (pos < CAND_CAP) {
          cand[(b * CAND_CAP + pos) * 2 + 0] = key;
          cand[(b * CAND_CAP + pos) * 2 + 1] = flat;
        }
      }
    }
  }
}

// ---- pass 4: per-batch bitonic sort of candidates, emit top-1000 ----
__global__ void __launch_bounds__(1024) sort_kernel(const uint32_t* cand, const uint32_t* candCnt,
                                                    float* topSc, uint32_t* topIdx) {
  __shared__ uint64_t sk[CAND_CAP];
  int b = blockIdx.x, tid = threadIdx.x;
  uint32_t cnt = candCnt[b]; if (cnt > CAND_CAP) cnt = CAND_CAP;
  const uint32_t* cb = cand + b * CAND_CAP * 2;
  for (int i = tid; i < CAND_CAP; i += 1024) {
    uint64_t v = 0ull;
    if (i < (int)cnt) {
      uint32_t key = cb[2 * i], fl = cb[2 * i + 1];
      v = ((uint64_t)key << 32) | (uint32_t)(~fl);  // desc score, asc index ties
    }
    sk[i] = v;
  }
  __syncthreads();
  for (int k = 2; k <= CAND_CAP; k <<= 1) {
    for (int j = k >> 1; j > 0; j >>= 1) {
      for (int i = tid; i < CAND_CAP; i += 1024) {
        int pr = i ^ j;
        if (pr > i) {
          bool desc = ((i & k) == 0);
          uint64_t x = sk[i], y = sk[pr];
          if (desc ? (x < y) : (x > y)) { sk[i] = y; sk[pr] = x; }
        }
      }
      __syncthreads();
    }
  }
  for (int k = tid; k < TOPK; k += 1024) {
    uint64_t v = sk[k];
    uint32_t key = (uint32_t)(v >> 32);
    uint32_t fl  = ~((uint32_t)v);
    float logit  = keyToFloat(key);
    topSc[b * TOPK + k]  = 1.0f / (1.0f + __expf(-logit));
    topIdx[b * TOPK + k] = fl;
  }
}

// ---- decode boxes for the 8000 selected entries ----
__global__ void decode_kernel(const float* r0, const float* r1, const float* r2,
                              const float* r3, const float* r4, const float* anchors,
                              const uint32_t* topIdx, float* boxesW, int* labelsW) {
  int idx = blockIdx.x * blockDim.x + threadIdx.x;
  if (idx >= BATCH * TOPK) return;
  int b = idx / TOPK;
  uint32_t flat = topIdx[idx];
  int anchor = (int)(flat / C_CLS);
  int label  = (int)(flat % C_CLS);
  const float* rp; int s, off;
  if      (anchor < 57600) { rp = r0; s = 6400; off = 0;     }
  else if (anchor < 72000) { rp = r1; s = 1600; off = 57600; }
  else if (anchor < 75600) { rp = r2; s = 400;  off = 72000; }
  else if (anchor < 76500) { rp = r3; s = 100;  off = 75600; }
  else                     { rp = r4; s = 25;   off = 76500; }
  int rr = anchor - off;
  int sp = rr / A_NUM, a = rr % A_NUM;
  const float* rb = rp + (size_t)b * (A_NUM * 4) * s;
  float dx = rb[(a * 4 + 0) * s + sp];
  float dy = rb[(a * 4 + 1) * s + sp];
  float dw = rb[(a * 4 + 2) * s + sp];
  float dh = rb[(a * 4 + 3) * s + sp];
  const float* an = anchors + (size_t)anchor * 4;
  float x1 = an[0], y1 = an[1], x2 = an[2], y2 = an[3];
  float w = x2 - x1, h = y2 - y1;
  float cx = x1 + 0.5f * w, cy = y1 + 0.5f * h;
  float pcx = dx * w + cx, pcy = dy * h + cy;
  float pw = expf(dw) * w, ph = expf(dh) * h;
  float* ob = boxesW + (size_t)idx * 4;
  ob[0] = pcx - 0.5f * pw; ob[1] = pcy - 0.5f * ph;
  ob[2] = pcx + 0.5f * pw; ob[3] = pcy + 0.5f * ph;
  labelsW[idx] = label;
}

// ---- suppression bitmask: one wave32 per 16x16 pair tile ----
// Union base area[i]+area[j] is computed on the matrix unit via
// V_WMMA_F32_16X16X4_F32 with A=[area_i | 1], B=[1 ; area_j].
__global__ void __launch_bounds__(256) mask_kernel(const float* boxesW, const int* labelsW,
                                                   uint32_t* mask) {
  __shared__ float sI[8][5][16];   // x1,y1,x2,y2,area (class-offset boxes), i-set
  __shared__ float sJ[8][5][16];   // j-set
  int w = threadIdx.x >> 5, lane = threadIdx.x & 31;
  int b = blockIdx.y;
  int t = blockIdx.x * 8 + w;      // grid sized so t < 2080 always
  int tr = 0, rem = t;
  while (rem >= (NTILE - tr)) { rem -= (NTILE - tr); tr++; }
  int tc = tr + rem;

  if (lane < 16) {
    int gi = tr * 16 + lane;
    float x1 = 0.f, y1 = 0.f, x2 = 0.f, y2 = 0.f;
    if (gi < TOPK) {
      const float* bp = boxesW + (size_t)(b * TOPK + gi) * 4;
      float off = (float)labelsW[b * TOPK + gi] * CLS_OFFSET;
      x1 = bp[0] + off; y1 = bp[1] + off; x2 = bp[2] + off; y2 = bp[3] + off;
    }
    sI[w][0][lane] = x1; sI[w][1][lane] = y1; sI[w][2][lane] = x2; sI[w][3][lane] = y2;
    sI[w][4][lane] = fmaxf(x2 - x1, 0.f) * fmaxf(y2 - y1, 0.f);
  } else {
    int l = lane - 16;
    int gj = tc * 16 + l;
    float x1 = 0.f, y1 = 0.f, x2 = 0.f, y2 = 0.f;
    if (gj < TOPK) {
      const float* bp = boxesW + (size_t)(b * TOPK + gj) * 4;
      float off = (float)labelsW[b * TOPK + gj] * CLS_OFFSET;
      x1 = bp[0] + off; y1 = bp[1] + off; x2 = bp[2] + off; y2 = bp[3] + off;
    }
    sJ[w][0][l] = x1; sJ[w][1][l] = y1; sJ[w][2][l] = x2; sJ[w][3][l] = y2;
    sJ[w][4][l] = fmaxf(x2 - x1, 0.f) * fmaxf(y2 - y1, 0.f);
  }
  __syncthreads();

  // A (16x4 f32): lanes0-15 {K0=area_i[M], K1=1}, lanes16-31 {K2=0, K3=0}
  // B (4x16 f32): lanes0-15 {K0=1, K1=area_j[N]}, upper lanes zero
  v2f av = {0.f, 0.f}, bv = {0.f, 0.f};
  if (lane < 16) {
    av.x = sI[w][4][lane]; av.y = 1.0f;
    bv.x = 1.0f;           bv.y = sJ[w][4][lane];
  }
  v8f c = {};
  c = __builtin_amdgcn_wmma_f32_16x16x4_f32(false, av, false, bv,
                                            (short)0, c, false, false);
  // c[v] for lane L = area_i[v + 8*(L>=16)] + area_j[L&15]
  int N = lane & 15;
  int baseM = (lane >> 4) << 3;
  float jx1 = sJ[w][0][N], jy1 = sJ[w][1][N], jx2 = sJ[w][2][N], jy2 = sJ[w][3][N];
  for (int v = 0; v < 8; v++) {
    int M = v + baseM;
    float ix1 = fmaxf(sI[w][0][M], jx1);
    float iy1 = fmaxf(sI[w][1][M], jy1);
    float ix2 = fminf(sI[w][2][M], jx2);
    float iy2 = fminf(sI[w][3][M], jy2);
    float inter = fmaxf(ix2 - ix1, 0.f) * fmaxf(iy2 - iy1, 0.f);
    float uni = c[v] - inter;                 // WMMA-produced area sum
    float iou = inter / fmaxf(uni, 1e-8f);
    int gi = tr * 16 + M, gj = tc * 16 + N;
    bool pred = (iou > IOU_THR) && (gj > gi) && (gi < TOPK) && (gj < TOPK);
    uint32_t bits = (uint32_t)__ballot(pred);  // lanes0-15: row v; lanes16-31: row v+8
    if (lane == 0) {
      uint32_t lo = bits & 0xFFFFu, hi = bits >> 16;
      int word = tc >> 1, sh = (tc & 1) * 16;
      if (lo) atomicOr(&mask[((size_t)b * PAD + tr * 16 + v    ) * 32 + word], lo << sh);
      if (hi) atomicOr(&mask[((size_t)b * PAD + tr * 16 + v + 8) * 32 + word], hi << sh);
    }
  }
}

// ---- serial NMS reduce + emit top-100: one wave32 per batch ----
__global__ void nms_kernel(const float* topSc, const uint32_t* mask,
                           const float* boxesW, const int* labelsW, float* out) {
  __shared__ uint32_t keepW[32];
  int b = blockIdx.x, lane = threadIdx.x;   // blockDim = 32
  uint32_t keep = 0;
  for (int q = 0; q < 32; q++) {
    int id = lane * 32 + q;
    bool v = (id < TOPK) && (topSc[b * TOPK + id] > CONF_THR);
    keep |= (v ? 1u : 0u) << q;
  }
  const uint32_t* mrow = mask + (size_t)b * PAD * 32;
  for (int i = 0; i < TOPK; i++) {
    uint32_t wv = __shfl(keep, i >> 5, 32);
    bool alive = (wv >> (i & 31)) & 1u;       // keep ⊆ valid, so keep[i] == alive
    if (alive) keep &= ~mrow[(size_t)i * 32 + lane];
  }
  keepW[lane] = keep;
  __syncthreads();
  if (lane == 0) {
    float* outLab = out;
    float* outSc  = out + BATCH * MAXOUT;
    float* outBox = out + 2 * BATCH * MAXOUT;
    int cnt = 0;
    for (int i = 0; i < TOPK && cnt < MAXOUT; i++) {      // kept, score-desc order
      if ((keepW[i >> 5] >> (i & 31)) & 1u) {
        outLab[b * MAXOUT + cnt] = (float)labelsW[b * TOPK + i];
        outSc [b * MAXOUT + cnt] = topSc[b * TOPK + i];
        const float* bp = boxesW + (size_t)(b * TOPK + i) * 4;
        float* ob = outBox + (size_t)(b * MAXOUT + cnt) * 4;
        ob[0] = bp[0]; ob[1] = bp[1]; ob[2] = bp[2]; ob[3] = bp[3];
        cnt++;
      }
    }
    for (int i = 0; i < TOPK && cnt < MAXOUT; i++) {      // zero-score filler
      if (!((keepW[i >> 5] >> (i & 31)) & 1u)) {
        outLab[b * MAXOUT + cnt] = -1.0f;
        outSc [b * MAXOUT + cnt] = 0.0f;
        const float* bp = boxesW + (size_t)(b * TOPK + i) * 4;
        float* ob = outBox + (size_t)(b * MAXOUT + cnt) * 4;
        ob[0] = bp[0]; ob[1] = bp[1]; ob[2] = bp[2]; ob[3] = bp[3];
        cnt++;
      }
    }
  }
}

extern "C" void kernel_launch(void* const* d_in, const int* in_sizes, int n_in,
                              void* d_out, int out_size, void* d_ws, size_t ws_size,
                              hipStream_t stream) {
  // input order: logits0, regress0, logits1, regress1, ..., logits4, regress4, anchors
  const float* L[5] = { (const float*)d_in[0], (const float*)d_in[2], (const float*)d_in[4],
                        (const float*)d_in[6], (const float*)d_in[8] };
  const float* R[5] = { (const float*)d_in[1], (const float*)d_in[3], (const float*)d_in[5],
                        (const float*)d_in[7], (const float*)d_in[9] };
  const float* anchors = (const float*)d_in[10];

  uint8_t*  ws      = (uint8_t*)d_ws;
  uint32_t* hist    = (uint32_t*)(ws + WS_HIST);
  uint32_t* mask    = (uint32_t*)(ws + WS_MASK);
  uint32_t* candCnt = (uint32_t*)(ws + WS_CCNT);
  uint32_t* thrKey  = (uint32_t*)(ws + WS_THR);
  uint32_t* cand    = (uint32_t*)(ws + WS_CAND);
  float*    topSc   = (float*)   (ws + WS_TSC);
  uint32_t* topIdx  = (uint32_t*)(ws + WS_TIDX);
  float*    boxesW  = (float*)   (ws + WS_BOX);
  int*      labelsW = (int*)     (ws + WS_LAB);

  zero_kernel<<<512, 256, 0, stream>>>((uint32_t*)ws, WS_ZERO_WORDS);
  hist_kernel<<<dim3(512, BATCH), 256, 0, stream>>>(L[0], L[1], L[2], L[3], L[4], hist);
  scan_kernel<<<1, 32, 0, stream>>>(hist, thrKey);
  collect_kernel<<<dim3(512, BATCH), 256, 0, stream>>>(L[0], L[1], L[2], L[3], L[4],
                                                       thrKey, candCnt, cand);
  sort_kernel<<<BATCH, 1024, 0, stream>>>(cand, candCnt, topSc, topIdx);
  decode_kernel<<<(BATCH * TOPK + 255) / 256, 256, 0, stream>>>(
      R[0], R[1], R[2], R[3], R[4], anchors, topIdx, boxesW, labelsW);
  mask_kernel<<<dim3(NPAIRTILES / 8, BATCH), 256, 0, stream>>>(boxesW, labelsW, mask);
  nms_kernel<<<BATCH, 32, 0, stream>>>(topSc, mask, boxesW, labelsW, (float*)d_out);
}